// MOELoadBalancingLoss_57621281243501
// MI455X (gfx1250) — compile-verified
//
#include <hip/hip_runtime.h>

typedef float v2f __attribute__((ext_vector_type(2)));
typedef float v8f __attribute__((ext_vector_type(8)));

#define NEXP   64
#define PB     1024   // probability-reduction blocks (8 waves each)
#define CB     128    // histogram blocks
#define ALPHA_ 0.01f

// Load the B operand (4x16 fp32 tile) for V_WMMA_F32_16X16X4_F32.
// Lane layout: lanes 0-15 cover rows {r, r+2}, lanes 16-31 cover rows {r+1, r+3},
// column = g*16 + (lane&15). Any K-row permutation is fine since we only sum over K.
__device__ __forceinline__ v2f load_b_tile(const float* __restrict__ probs,
                                           int r, int half, int col) {
    v2f b;
    b.x = probs[(size_t)(r + half) * NEXP + col];
    b.y = probs[(size_t)(r + 2 + half) * NEXP + col];
    return b;
}

__global__ __launch_bounds__(256) void moe_main(const float* __restrict__ probs,
                                                const int*   __restrict__ idx,
                                                float*       __restrict__ wsf,  // PB*64 colsum partials
                                                unsigned*    __restrict__ wsc,  // CB*64 count partials
                                                int rows, int nIdx) {
    __shared__ float    lsum[NEXP];
    __shared__ unsigned lhist[NEXP];
    const int tid = threadIdx.x;

    if (blockIdx.x < PB) {
        // ---- column-sum role: colsum[n] = ones(1xB) . P(Bx64) via WMMA ----
        v2f a; a.x = 1.0f; a.y = 1.0f;      // A = all-ones 16x4
        v8f c = {};                          // fp32 accumulator (16x16)
        const int lane    = tid & 31;
        const int wib     = tid >> 5;                 // wave in block, 0..7
        const int pw      = blockIdx.x * 8 + wib;     // global prob-wave id
        const int g       = pw & 3;                   // 16-column group
        const int colBase = g * 16;
        const int half    = lane >> 4;
        const int col     = colBase + (lane & 15);
        const int tiles   = rows >> 2;                // 4 rows per tile
        const int stride  = PB * 2;                   // tile-workers per group

        int t = pw >> 2;
        for (; t + 3 * stride < tiles; t += 4 * stride) {
            // 8 loads issued before the dependent WMMA chain -> deep MLP
            v2f b0 = load_b_tile(probs, (t             ) << 2, half, col);
            v2f b1 = load_b_tile(probs, (t +     stride) << 2, half, col);
            v2f b2 = load_b_tile(probs, (t + 2 * stride) << 2, half, col);
            v2f b3 = load_b_tile(probs, (t + 3 * stride) << 2, half, col);
            c = __builtin_amdgcn_wmma_f32_16x16x4_f32(false, a, false, b0, (short)0, c, false, false);
            c = __builtin_amdgcn_wmma_f32_16x16x4_f32(false, a, false, b1, (short)0, c, false, false);
            c = __builtin_amdgcn_wmma_f32_16x16x4_f32(false, a, false, b2, (short)0, c, false, false);
            c = __builtin_amdgcn_wmma_f32_16x16x4_f32(false, a, false, b3, (short)0, c, false, false);
        }
        for (; t < tiles; t += stride) {
            v2f b = load_b_tile(probs, t << 2, half, col);
            c = __builtin_amdgcn_wmma_f32_16x16x4_f32(false, a, false, b, (short)0, c, false, false);
        }

        // D row M=0 lives in c[0], lanes 0-15 (N = lane). Combine 8 waves via LDS.
        if (tid < NEXP) lsum[tid] = 0.0f;
        __syncthreads();
        if (lane < 16) atomicAdd(&lsum[colBase + lane], c[0]);
        __syncthreads();
        if (tid < NEXP) wsf[(size_t)blockIdx.x * NEXP + tid] = lsum[tid];
    } else {
        // ---- histogram role ----
        const int cb = blockIdx.x - PB;
        if (tid < NEXP) lhist[tid] = 0u;
        __syncthreads();
        const int   n4   = nIdx >> 2;
        const int4* idx4 = (const int4*)idx;
        for (int i = cb * 256 + tid; i < n4; i += CB * 256) {
            int4 v = idx4[i];
            atomicAdd(&lhist[v.x], 1u);
            atomicAdd(&lhist[v.y], 1u);
            atomicAdd(&lhist[v.z], 1u);
            atomicAdd(&lhist[v.w], 1u);
        }
        __syncthreads();
        if (tid < NEXP) wsc[(size_t)cb * NEXP + tid] = lhist[tid];
    }
}

__global__ __launch_bounds__(256) void moe_finalize(const float* __restrict__ wsf,
                                                    const unsigned* __restrict__ wsc,
                                                    float* __restrict__ out,
                                                    int rows, int nIdx) {
    __shared__ float red[256];
    __shared__ float colsum[NEXP];
    __shared__ float cnt[NEXP];
    const int tid  = threadIdx.x;
    const int e    = tid & 63;
    const int part = tid >> 6;   // 0..3

    float s = 0.0f;
    for (int b = part; b < PB; b += 4) s += wsf[(size_t)b * NEXP + e];
    red[tid] = s;
    __syncthreads();
    if (part == 0) colsum[e] = red[e] + red[e + 64] + red[e + 128] + red[e + 192];
    __syncthreads();

    float cs = 0.0f;
    for (int b = part; b < CB; b += 4) cs += (float)wsc[(size_t)b * NEXP + e];
    red[tid] = cs;
    __syncthreads();
    if (part == 0) cnt[e] = red[e] + red[e + 64] + red[e + 128] + red[e + 192];
    __syncthreads();

    if (tid < NEXP) red[tid] = colsum[tid] * cnt[tid];
    __syncthreads();
    if (tid == 0) {
        float tot = 0.0f;
        for (int i = 0; i < NEXP; ++i) tot += red[i];
        // loss = ALPHA * NEXP * sum_e (colsum_e/rows) * (cnt_e/(rows*top_k))
        out[0] = ALPHA_ * (float)NEXP * tot / ((float)rows * (float)nIdx);
    }
}

extern "C" void kernel_launch(void* const* d_in, const int* in_sizes, int n_in,
                              void* d_out, int out_size, void* d_ws, size_t ws_size,
                              hipStream_t stream) {
    const float* probs = (const float*)d_in[0];
    const int*   idx   = (const int*)d_in[1];
    const int rows = in_sizes[0] / NEXP;   // 1048576
    const int nIdx = in_sizes[1];          // 8388608

    float*    wsf = (float*)d_ws;
    unsigned* wsc = (unsigned*)((char*)d_ws + (size_t)PB * NEXP * sizeof(float));

    moe_main<<<PB + CB, 256, 0, stream>>>(probs, idx, wsf, wsc, rows, nIdx);
    moe_finalize<<<1, 256, 0, stream>>>(wsf, wsc, (float*)d_out, rows, nIdx);
}